// PostNet_89859305767166
// MI455X (gfx1250) — compile-verified
//
#include <hip/hip_runtime.h>
#include <stdint.h>

// Problem constants (fixed by the reference).
#define N_VIEWS 10
#define BB 2
#define HH 512
#define WW 640
#define HW (HH * WW)
#define TPB 256

// Precomputed per-batch / per-(view,batch) camera constants. Computed once per
// launch by a 32-thread setup kernel. The main kernel receives this table via a
// `const * __restrict__` KERNEL ARGUMENT (not a module global): noalias args
// survive the inline-asm "memory" clobber, so these loads lower to clause'd
// s_load_* into SGPRs that feed v_fma directly (no per-lane VMEM, no VGPRs).
struct GeoConsts {
    float M[BB][9];            // ref_R^T @ inv(ref_K)
    float refc[BB][3];         // -(ref_R^T @ ref_t)
    float Rv[N_VIEWS][BB][9];  // view rotation (row major)
    float qo[N_VIEWS][BB][3];  // -(Rv @ view_c): q = Rv@g + qo
    float Kv[N_VIEWS][BB][6];  // rows 0,1 of view K
    float P[N_VIEWS][BB][9];   // (ref_K @ ref_R) @ Rv^T
    float o[N_VIEWS][BB][3];   // (ref_K @ ref_R) @ (view_c - ref_c)
};
__device__ GeoConsts g_consts;

__device__ __forceinline__ void mat3_inv(const float* a, float* inv) {
    // General 3x3 inverse via adjugate (matches jnp.linalg.inv semantics).
    float c00 = a[4] * a[8] - a[5] * a[7];
    float c01 = a[5] * a[6] - a[3] * a[8];
    float c02 = a[3] * a[7] - a[4] * a[6];
    float det = a[0] * c00 + a[1] * c01 + a[2] * c02;
    float id  = 1.0f / det;
    inv[0] = c00 * id;
    inv[1] = (a[2] * a[7] - a[1] * a[8]) * id;
    inv[2] = (a[1] * a[5] - a[2] * a[4]) * id;
    inv[3] = c01 * id;
    inv[4] = (a[0] * a[8] - a[2] * a[6]) * id;
    inv[5] = (a[2] * a[3] - a[0] * a[5]) * id;
    inv[6] = c02 * id;
    inv[7] = (a[1] * a[6] - a[0] * a[7]) * id;
    inv[8] = (a[0] * a[4] - a[1] * a[3]) * id;
}

__device__ __forceinline__ void load_cam(const float* cam44, float* R3, float* t3,
                                         const float* k44, float* K3) {
#pragma unroll
    for (int i = 0; i < 3; ++i) {
#pragma unroll
        for (int j = 0; j < 3; ++j) {
            R3[i * 3 + j] = cam44[i * 4 + j];
            K3[i * 3 + j] = k44[i * 4 + j];
        }
        t3[i] = cam44[i * 4 + 3];
    }
}

__global__ void geo_setup_kernel(GeoConsts* __restrict__ C,
                                 const float* __restrict__ ref_cams,
                                 const float* __restrict__ view_cams) {
    const int tid = threadIdx.x;

    // Per-batch constants (threads 0..BB-1).
    if (tid < BB) {
        const int b = tid;
        float R[9], K[9], t[3], invK[9];
        load_cam(ref_cams + (size_t)b * 32, R, t, ref_cams + (size_t)b * 32 + 16, K);
        mat3_inv(K, invK);
#pragma unroll
        for (int i = 0; i < 3; ++i) {
#pragma unroll
            for (int j = 0; j < 3; ++j)
                C->M[b][i * 3 + j] = R[0 * 3 + i] * invK[0 * 3 + j] +
                                     R[1 * 3 + i] * invK[1 * 3 + j] +
                                     R[2 * 3 + i] * invK[2 * 3 + j];
            C->refc[b][i] = -(R[0 * 3 + i] * t[0] + R[1 * 3 + i] * t[1] +
                              R[2 * 3 + i] * t[2]);
        }
    }

    // Per-(view,batch) constants (threads 0..N_VIEWS*BB-1, each independent).
    if (tid < N_VIEWS * BB) {
        const int v = tid / BB;
        const int b = tid % BB;

        // Recompute ref-side constants locally (no cross-thread dependency).
        float R[9], K[9], t[3];
        load_cam(ref_cams + (size_t)b * 32, R, t, ref_cams + (size_t)b * 32 + 16, K);
        float refc[3], KR[9];
#pragma unroll
        for (int i = 0; i < 3; ++i) {
#pragma unroll
            for (int j = 0; j < 3; ++j)
                KR[i * 3 + j] = K[i * 3 + 0] * R[0 * 3 + j] +
                                K[i * 3 + 1] * R[1 * 3 + j] +
                                K[i * 3 + 2] * R[2 * 3 + j];
            refc[i] = -(R[0 * 3 + i] * t[0] + R[1 * 3 + i] * t[1] + R[2 * 3 + i] * t[2]);
        }

        const float* Ev = view_cams + ((size_t)v * BB + b) * 32;
        float Rv[9], Kv3[9], tv[3];
        load_cam(Ev, Rv, tv, Ev + 16, Kv3);

        // view_c = -(Rv^T @ tv)
        float c0 = -(Rv[0] * tv[0] + Rv[3] * tv[1] + Rv[6] * tv[2]);
        float c1 = -(Rv[1] * tv[0] + Rv[4] * tv[1] + Rv[7] * tv[2]);
        float c2 = -(Rv[2] * tv[0] + Rv[5] * tv[1] + Rv[8] * tv[2]);

#pragma unroll
        for (int i = 0; i < 9; ++i) C->Rv[v][b][i] = Rv[i];
#pragma unroll
        for (int i = 0; i < 6; ++i) C->Kv[v][b][i] = Kv3[i];

        // qo = -(Rv @ view_c) so per-pixel q = Rv@g + qo  (== Rv@(g - view_c))
        C->qo[v][b][0] = -(Rv[0] * c0 + Rv[1] * c1 + Rv[2] * c2);
        C->qo[v][b][1] = -(Rv[3] * c0 + Rv[4] * c1 + Rv[5] * c2);
        C->qo[v][b][2] = -(Rv[6] * c0 + Rv[7] * c1 + Rv[8] * c2);

        // P = KR @ Rv^T ;  o = KR @ (view_c - ref_c)
        const float d0 = c0 - refc[0], d1 = c1 - refc[1], d2 = c2 - refc[2];
#pragma unroll
        for (int i = 0; i < 3; ++i) {
#pragma unroll
            for (int j = 0; j < 3; ++j)
                C->P[v][b][i * 3 + j] = KR[i * 3 + 0] * Rv[j * 3 + 0] +
                                        KR[i * 3 + 1] * Rv[j * 3 + 1] +
                                        KR[i * 3 + 2] * Rv[j * 3 + 2];
            C->o[v][b][i] = KR[i * 3 + 0] * d0 + KR[i * 3 + 1] * d1 +
                            KR[i * 3 + 2] * d2;
        }
    }
}

__global__ __launch_bounds__(TPB) void geo_consistency_kernel(
    const float* __restrict__ ref_depth,    // (B,1,H,W)
    const float* __restrict__ view_depth,   // (N,B,H,W)
    const GeoConsts* __restrict__ C,        // precomputed camera constants
    float* __restrict__ out)                // [mask*depth | mask], each (B,1,H,W)
{
    __shared__ float s_rd[TPB];

    const int b = blockIdx.y;
    const int p = blockIdx.x * TPB + threadIdx.x;   // pixel index within batch
    const int x = p % WW;
    const int y = p / WW;
    const float xf = (float)x, yf = (float)y;

    // ---- CDNA5 async data mover: stage this block's ref-depth tile into LDS.
    // Each lane reads back only the slot its own async op filled, so the
    // wave-scope s_wait_asynccnt is the complete dependency (no barrier).
#if defined(__HIP_DEVICE_COMPILE__)
    {
        uint32_t lds_off = (uint32_t)(uintptr_t)(&s_rd[threadIdx.x]);
        uint64_t gaddr   = (uint64_t)(uintptr_t)(ref_depth + (size_t)b * HW + p);
        asm volatile("global_load_async_to_lds_b32 %0, %1, off"
                     :: "v"(lds_off), "v"(gaddr) : "memory");
        asm volatile("s_wait_asynccnt 0x0" ::: "memory");
    }
#else
    s_rd[threadIdx.x] = ref_depth[(size_t)b * HW + p];
#endif
    const float rd = s_rd[threadIdx.x];

    // Back-project ref pixel to world (view-independent): g = M@(x,y,1)*rd + refc
    const float vx = xf * rd, vy = yf * rd, vz = rd;
    const float* M    = C->M[b];
    const float* refc = C->refc[b];
    const float g0 = fmaf(M[0], vx, fmaf(M[1], vy, fmaf(M[2], vz, refc[0])));
    const float g1 = fmaf(M[3], vx, fmaf(M[4], vy, fmaf(M[5], vz, refc[1])));
    const float g2 = fmaf(M[6], vx, fmaf(M[7], vy, fmaf(M[8], vz, refc[2])));

    const float* vdm_b = view_depth + (size_t)b * HW;   // view stride = BB*HW

    int count = 0;
#pragma unroll
    for (int v = 0; v < N_VIEWS; ++v) {
        const float* vdm = vdm_b + (size_t)v * (size_t)BB * HW;
        // global_prefetch_b8: pre-touch next view's depth neighborhood
        // (the data-dependent gather lands within a few pixels of p).
        if (v + 1 < N_VIEWS)
            __builtin_prefetch(vdm_b + (size_t)(v + 1) * (size_t)BB * HW + p, 0, 0);

        const float* Rv = C->Rv[v][b];
        const float* qo = C->qo[v][b];
        const float* Kv = C->Kv[v][b];
        const float* P  = C->P[v][b];
        const float* o  = C->o[v][b];

        // view-local coords: q = Rv@g + qo
        const float q0 = fmaf(Rv[0], g0, fmaf(Rv[1], g1, fmaf(Rv[2], g2, qo[0])));
        const float q1 = fmaf(Rv[3], g0, fmaf(Rv[4], g1, fmaf(Rv[5], g2, qo[1])));
        const float q2 = fmaf(Rv[6], g0, fmaf(Rv[7], g1, fmaf(Rv[8], g2, qo[2])));

        const float z  = q2;
        const float rz = 1.0f / z;          // the one unavoidable IEEE divide
        const float lx = q0 * rz;
        const float ly = q1 * rz;
        // lz == 1 exactly for any finite nonzero z; in the z==0/NaN case every
        // consistency predicate below is already false, so lz := 1 is
        // output-equivalent to the reference's q2/z.

        // pixel coords in view: vc = Kv @ (lx,ly,1)
        const float pcx = fmaf(Kv[0], lx, fmaf(Kv[1], ly, Kv[2]));
        const float pcy = fmaf(Kv[3], lx, fmaf(Kv[4], ly, Kv[5]));

        // NaN-safe in-bounds test (NaN compares false -> oob, like reference)
        const bool inb = (pcy >= 0.0f) & (pcy < (float)HH) &
                         (pcx >= 0.0f) & (pcx < (float)WW) & (z > 0.0f);
        const int idx = inb ? ((int)pcy * WW + (int)pcx) : 0;  // trunc toward 0

        float gd = vdm[idx];
        if (!inb) gd = 1e-7f;

        // n = KR @ (Rv^T@(vlc*gd) + view_c - refc) = gd*(P@vlc) + o
        const float t0 = fmaf(P[0], lx, fmaf(P[1], ly, P[2]));
        const float t1 = fmaf(P[3], lx, fmaf(P[4], ly, P[5]));
        const float t2 = fmaf(P[6], lx, fmaf(P[7], ly, P[8]));
        const float n0 = fmaf(gd, t0, o[0]);
        const float n1 = fmaf(gd, t1, o[1]);
        const float n2 = fmaf(gd, t2, o[2]);

        // dist < 1  <=>  (n0/n2 - x)^2 + (n1/n2 - y)^2 < 1
        //           <=>  (n0 - x*n2)^2 + (n1 - y*n2)^2 < n2^2
        // (divide both sides by n2^2 > 0; n2==0 -> both false; NaN -> false).
        // Eliminates the second IEEE divide entirely.
        const float e0 = fmaf(-xf, n2, n0);
        const float e1 = fmaf(-yf, n2, n1);
        const float lhs = fmaf(e0, e0, e1 * e1);
        const bool condD = lhs < n2 * n2;

        // |z1-rd|*2/|rd+z1| < 0.01  <=>  |z1-rd|*2 < 0.01*|rd+z1|
        // (both sides non-negative; NaN -> false either way)
        const float z1 = fmaxf(n2, 0.0f);
        const bool condR = fabsf(z1 - rd) * 2.0f < 0.01f * fabsf(rd + z1);

        const bool cons = condD & condR & (z1 > 0.0f);
        count += cons ? 1 : 0;
    }

    // (rd > 0) is view-invariant: hoisted out of the loop (if rd <= 0 every
    // per-view predicate in the reference is false, so count==0 either way).
    const float mask = ((count >= 3) & (rd > 0.0f)) ? 1.0f : 0.0f;
    out[(size_t)b * HW + p]                   = mask * rd;   // masked depth
    out[(size_t)BB * HW + (size_t)b * HW + p] = mask;        // mask
}

extern "C" void kernel_launch(void* const* d_in, const int* in_sizes, int n_in,
                              void* d_out, int out_size, void* d_ws, size_t ws_size,
                              hipStream_t stream) {
    // setup_inputs order: 0=ref_color_maps (unused), 1=view_color_maps (unused),
    // 2=ref_depth_map, 3=view_depth_maps, 4=ref_cams, 5=view_cams
    (void)in_sizes; (void)n_in; (void)out_size; (void)d_ws; (void)ws_size;
    const float* ref_depth  = (const float*)d_in[2];
    const float* view_depth = (const float*)d_in[3];
    const float* ref_cams   = (const float*)d_in[4];
    const float* view_cams  = (const float*)d_in[5];
    float* out = (float*)d_out;

    // Device address of the constants table (module query; graph-capture safe).
    void* consts_ptr = nullptr;
    (void)hipGetSymbolAddress(&consts_ptr, HIP_SYMBOL(g_consts));
    GeoConsts* Cd = (GeoConsts*)consts_ptr;

    // Tiny setup kernel: fold all uniform camera math (incl. 3x3 inverse and
    // the KR@Rv^T / KR@(view_c-ref_c) fusions) into the constants table.
    geo_setup_kernel<<<1, 32, 0, stream>>>(Cd, ref_cams, view_cams);

    dim3 grid(HW / TPB, BB);   // 1280 x 2 blocks, 8 wave32 per block
    dim3 block(TPB);
    geo_consistency_kernel<<<grid, block, 0, stream>>>(ref_depth, view_depth, Cd, out);
}